// HierarchicalPooling_68650757259333
// MI455X (gfx1250) — compile-verified
//
#include <hip/hip_runtime.h>
#include <math.h>

typedef __attribute__((ext_vector_type(2))) float v2f;
typedef __attribute__((ext_vector_type(8))) float v8f;

#define N_PTS 100000
#define C_DIM 128
#define K_DIM 32
#define B_SEG 64
#define TAU_INV 1.0f
#define EPS_F 1e-9f

#define N_TILES (N_PTS / 16)                 // 6250 row tiles
#define OUT_ELEMS (B_SEG * K_DIM * C_DIM)    // 262144
#define S_ELEMS (N_PTS * K_DIM)              // 3200000

// ---------------------------------------------------------------------------
// ws layout (floats): ws[0] = sum over (n,k) of s*log(s+eps); ws[1..32] = col sums of s
__global__ void hp_init_ws(float* __restrict__ ws) {
    int i = threadIdx.x;
    if (i < 33) ws[i] = 0.0f;
}

// ---------------------------------------------------------------------------
// Fused: h = relu(x@W1+b1); logits = h@W2+b2; s = softmax(logits+g); stats.
// One wave per 16-row tile. All matrix math through V_WMMA_F32_16X16X4_F32.
__global__ __launch_bounds__(128) void hp_assign_kernel(
    const float* __restrict__ x, const float* __restrict__ W1,
    const float* __restrict__ b1, const float* __restrict__ W2,
    const float* __restrict__ b2, const float* __restrict__ gumbel,
    float* __restrict__ s_out, float* __restrict__ ws)
{
    __shared__ float lds_h[4][16 * 132];     // 132-float row pitch: bank-conflict pad
    const int lane  = threadIdx.x & 31;
    const int wave  = threadIdx.x >> 5;
    const int tile  = blockIdx.x * 4 + wave;
    if (tile >= N_TILES) return;             // wave-uniform exit: EXEC stays all-1 for WMMA

    const int rowBase = tile * 16;
    const int mrow    = lane & 15;           // A row / B,C,D column within tile
    const int khalf   = (lane >> 4) * 2;     // lanes 16-31 hold k+2,k+3
    const int mhi     = (lane >> 4) << 3;    // C/D: lanes 16-31 hold rows 8-15

    // ---------------- GEMM1: acc[ct] = x_tile @ W1[:, ct*16:+16] ----------------
    v8f acc[8];
    #pragma unroll
    for (int t = 0; t < 8; ++t) acc[t] = (v8f){0.f,0.f,0.f,0.f,0.f,0.f,0.f,0.f};

    const float* __restrict__ xrow = x + (size_t)(rowBase + mrow) * C_DIM;
    #pragma unroll 4
    for (int kc = 0; kc < 32; ++kc) {
        const int kb = kc * 4 + khalf;
        v2f a;
        a.x = xrow[kb];
        a.y = xrow[kb + 1];
        #pragma unroll
        for (int ct = 0; ct < 8; ++ct) {
            const int col = ct * 16 + mrow;
            v2f bf;
            bf.x = W1[(size_t)kb * C_DIM + col];
            bf.y = W1[(size_t)(kb + 1) * C_DIM + col];
            acc[ct] = __builtin_amdgcn_wmma_f32_16x16x4_f32(
                false, a, false, bf, (short)0, acc[ct], false, false);
        }
    }

    // bias + relu, stage h tile (row-major) into LDS for the next A-operand
    float* __restrict__ hl = lds_h[wave];
    #pragma unroll
    for (int ct = 0; ct < 8; ++ct) {
        const int col  = ct * 16 + mrow;
        const float bb = b1[col];
        #pragma unroll
        for (int v = 0; v < 8; ++v) {
            const int m = v + mhi;
            hl[m * 132 + col] = fmaxf(acc[ct][v] + bb, 0.0f);
        }
    }

    // ---------------- GEMM2: lg[ct] = h_tile @ W2[:, ct*16:+16] ----------------
    v8f lg[2];
    lg[0] = (v8f){0.f,0.f,0.f,0.f,0.f,0.f,0.f,0.f};
    lg[1] = lg[0];
    #pragma unroll 4
    for (int kc = 0; kc < 32; ++kc) {
        const int kb = kc * 4 + khalf;
        v2f a;
        a.x = hl[mrow * 132 + kb];
        a.y = hl[mrow * 132 + kb + 1];
        #pragma unroll
        for (int ct = 0; ct < 2; ++ct) {
            const int col = ct * 16 + mrow;
            v2f bf;
            bf.x = W2[(size_t)kb * K_DIM + col];
            bf.y = W2[(size_t)(kb + 1) * K_DIM + col];
            lg[ct] = __builtin_amdgcn_wmma_f32_16x16x4_f32(
                false, a, false, bf, (short)0, lg[ct], false, false);
        }
    }

    // ---------------- logits + gumbel, rowwise softmax over 32 cols ----------------
    float sv[2][8];
    #pragma unroll
    for (int ct = 0; ct < 2; ++ct) {
        const int col  = ct * 16 + mrow;
        const float bb = b2[col];
        #pragma unroll
        for (int v = 0; v < 8; ++v) {
            const int row = rowBase + v + mhi;
            sv[ct][v] = (lg[ct][v] + bb + gumbel[(size_t)row * K_DIM + col]) * TAU_INV;
        }
    }
    // Lanes 0-15 hold rows 0-7, lanes 16-31 rows 8-15: xor 1/2/4/8 reduces each row.
    #pragma unroll
    for (int v = 0; v < 8; ++v) {
        float mx = fmaxf(sv[0][v], sv[1][v]);
        mx = fmaxf(mx, __shfl_xor(mx, 1, 32));
        mx = fmaxf(mx, __shfl_xor(mx, 2, 32));
        mx = fmaxf(mx, __shfl_xor(mx, 4, 32));
        mx = fmaxf(mx, __shfl_xor(mx, 8, 32));
        float e0 = __expf(sv[0][v] - mx);
        float e1 = __expf(sv[1][v] - mx);
        float sm = e0 + e1;
        sm += __shfl_xor(sm, 1, 32);
        sm += __shfl_xor(sm, 2, 32);
        sm += __shfl_xor(sm, 4, 32);
        sm += __shfl_xor(sm, 8, 32);
        const float inv = 1.0f / sm;
        sv[0][v] = e0 * inv;
        sv[1][v] = e1 * inv;
    }

    // ---------------- write s, accumulate entropy and column sums ----------------
    float ent = 0.0f;
    float cs[2];
    #pragma unroll
    for (int ct = 0; ct < 2; ++ct) {
        const int col = ct * 16 + mrow;
        float c = 0.0f;
        #pragma unroll
        for (int v = 0; v < 8; ++v) {
            const int row = rowBase + v + mhi;
            const float sval = sv[ct][v];
            s_out[(size_t)row * K_DIM + col] = sval;
            ent += sval * __logf(sval + EPS_F);
            c += sval;
        }
        cs[ct] = c;
    }
    // wave-reduce entropy partial -> one atomic per wave
    ent += __shfl_xor(ent, 16, 32);
    ent += __shfl_xor(ent, 8, 32);
    ent += __shfl_xor(ent, 4, 32);
    ent += __shfl_xor(ent, 2, 32);
    ent += __shfl_xor(ent, 1, 32);
    if (lane == 0) atomicAdd(&ws[0], ent);
    // combine rows 0-7 with rows 8-15 (lane ^ 16) per column
    cs[0] += __shfl_xor(cs[0], 16, 32);
    cs[1] += __shfl_xor(cs[1], 16, 32);
    if (lane < 16) {
        atomicAdd(&ws[1 + lane], cs[0]);
        atomicAdd(&ws[1 + 16 + lane], cs[1]);
    }
}

// ---------------------------------------------------------------------------
// out[b] = s[seg]^T @ x[seg]; batch is sorted -> binary-search segment bounds.
// One wave per (b, ktile, ctile) output 16x16 tile. Main loop: 16 rows/iter
// across 4 independent WMMA accumulator chains (hides WMMA latency); guarded
// masked-tail loop handles the last (len & 15) rows.
__global__ __launch_bounds__(128) void hp_scatter_kernel(
    const float* __restrict__ x, const int* __restrict__ batch,
    const float* __restrict__ s, float* __restrict__ out)
{
    const int lane  = threadIdx.x & 31;
    const int w     = blockIdx.x * 4 + (threadIdx.x >> 5); // 0..1023
    const int b     = w >> 4;
    const int ktile = (w >> 3) & 1;
    const int ctile = w & 7;

    // lower_bound(batch, b) and lower_bound(batch, b+1)
    int lo = 0, hi = N_PTS;
    while (lo < hi) { int mid = (lo + hi) >> 1; if (batch[mid] < b) lo = mid + 1; else hi = mid; }
    const int start = lo;
    hi = N_PTS;
    while (lo < hi) { int mid = (lo + hi) >> 1; if (batch[mid] < b + 1) lo = mid + 1; else hi = mid; }
    const int end = lo;

    const int mrow  = lane & 15;
    const int khalf = (lane >> 4) * 2;
    const int kcol  = ktile * 16 + mrow;   // A row  (k index)
    const int ccol  = ctile * 16 + mrow;   // B col  (c index)

    const float* __restrict__ sc = s + kcol;   // column kcol of s
    const float* __restrict__ xc = x + ccol;   // column ccol of x

    v8f acc0 = (v8f){0.f,0.f,0.f,0.f,0.f,0.f,0.f,0.f};
    v8f acc1 = acc0, acc2 = acc0, acc3 = acc0;

    const int len   = end - start;
    const int nmain = start + (len & ~15);

    // ---- unguarded main loop: 16 segment rows per iteration, 4 WMMA chains ----
    for (int n0 = start; n0 < nmain; n0 += 16) {
        #pragma unroll
        for (int u = 0; u < 4; ++u) {
            const int na = n0 + u * 4 + khalf;
            v2f a, bf;
            a.x  = sc[(size_t)na * K_DIM];
            a.y  = sc[(size_t)(na + 1) * K_DIM];
            bf.x = xc[(size_t)na * C_DIM];
            bf.y = xc[(size_t)(na + 1) * C_DIM];
            if (u == 0)      acc0 = __builtin_amdgcn_wmma_f32_16x16x4_f32(false, a, false, bf, (short)0, acc0, false, false);
            else if (u == 1) acc1 = __builtin_amdgcn_wmma_f32_16x16x4_f32(false, a, false, bf, (short)0, acc1, false, false);
            else if (u == 2) acc2 = __builtin_amdgcn_wmma_f32_16x16x4_f32(false, a, false, bf, (short)0, acc2, false, false);
            else             acc3 = __builtin_amdgcn_wmma_f32_16x16x4_f32(false, a, false, bf, (short)0, acc3, false, false);
        }
    }

    // ---- masked tail: at most 4 iterations ----
    for (int n0 = nmain; n0 < end; n0 += 4) {
        const int na = n0 + khalf;
        v2f a, bf;
        a.x  = (na     < end) ? sc[(size_t)na * K_DIM]       : 0.0f;
        a.y  = (na + 1 < end) ? sc[(size_t)(na + 1) * K_DIM] : 0.0f;
        bf.x = (na     < end) ? xc[(size_t)na * C_DIM]       : 0.0f;
        bf.y = (na + 1 < end) ? xc[(size_t)(na + 1) * C_DIM] : 0.0f;
        acc0 = __builtin_amdgcn_wmma_f32_16x16x4_f32(false, a, false, bf, (short)0, acc0, false, false);
    }

    const v8f accT = (acc0 + acc1) + (acc2 + acc3);
    const int mhi = (lane >> 4) << 3;
    #pragma unroll
    for (int v = 0; v < 8; ++v) {
        const int k = ktile * 16 + v + mhi;
        out[((size_t)b * K_DIM + k) * C_DIM + ctile * 16 + mrow] = accT[v];
    }
}

// ---------------------------------------------------------------------------
__global__ void hp_loss_kernel(const float* __restrict__ ws, float* __restrict__ loss_out) {
    const int k = threadIdx.x;               // 32 threads
    float avg  = ws[1 + k] * (1.0f / (float)N_PTS);
    float term = avg * __logf(avg + EPS_F);
    term += __shfl_xor(term, 1, 32);
    term += __shfl_xor(term, 2, 32);
    term += __shfl_xor(term, 4, 32);
    term += __shfl_xor(term, 8, 32);
    term += __shfl_xor(term, 16, 32);
    if (k == 0) {
        const float entropy = -(ws[0] * (1.0f / (float)N_PTS));
        loss_out[0] = entropy + term;
    }
}

// ---------------------------------------------------------------------------
extern "C" void kernel_launch(void* const* d_in, const int* in_sizes, int n_in,
                              void* d_out, int out_size, void* d_ws, size_t ws_size,
                              hipStream_t stream) {
    const float* x      = (const float*)d_in[0];
    const int*   batch  = (const int*)  d_in[1];
    const float* W1     = (const float*)d_in[2];
    const float* b1     = (const float*)d_in[3];
    const float* W2     = (const float*)d_in[4];
    const float* b2     = (const float*)d_in[5];
    const float* gumbel = (const float*)d_in[6];

    float* out_seg = (float*)d_out;                    // (B, K, C)
    float* s_out   = out_seg + OUT_ELEMS;              // (N, K)
    float* loss    = s_out + S_ELEMS;                  // scalar
    float* ws      = (float*)d_ws;

    hp_init_ws<<<1, 64, 0, stream>>>(ws);

    const int assign_blocks = (N_TILES + 3) / 4;       // 4 waves (tiles) per block
    hp_assign_kernel<<<assign_blocks, 128, 0, stream>>>(
        x, W1, b1, W2, b2, gumbel, s_out, ws);

    hp_scatter_kernel<<<(B_SEG * 16) / 4, 128, 0, stream>>>(x, batch, s_out, out_seg);

    hp_loss_kernel<<<1, 32, 0, stream>>>(ws, loss);
}